// FeatureWarperSoftsplat_35854386987569
// MI455X (gfx1250) — compile-verified
//
#include <hip/hip_runtime.h>
#include <stdint.h>

// Problem shape (fixed by the reference)
#define NB 8
#define NC 128
#define NH 256
#define NW 256
#define HWPLANE (NH * NW)                                   // 65536
#define NUM_ELEMS ((size_t)NB * NC * NH * NW)               // 67,108,864 warped floats
#define DEN_ELEMS ((size_t)NB * NH * NW)                    // 524,288 metric floats
#define ROWS_PER_BLK 4

__device__ __constant__ const float kE   = 2.71828182845904523536f; // exp(metric=1)
__device__ __constant__ const float kEPS = 1e-7f;

typedef unsigned int u32x4 __attribute__((ext_vector_type(4)));
typedef int          i32x4 __attribute__((ext_vector_type(4)));
typedef int          i32x8 __attribute__((ext_vector_type(8)));

// CDNA5 aperture rule: for LDS generic pointers, addr[31:0] IS the LDS byte offset.
__device__ __forceinline__ uint32_t lds_byte_addr(const void* p) {
  return (uint32_t)(uintptr_t)p;
}

// One Tensor-Data-Mover transfer of a 2D tile (tile_d1 rows of tile_d0 fp32
// elements, rows separated by stride_d0 elements) from global into LDS.
// Wave-level op: issue from a single wave, s_wait_tensorcnt, then barrier.
__device__ __forceinline__ void tdm_load_rows_to_lds(
    uint32_t lds_addr, const void* gptr,
    uint32_t tensor_d0, uint32_t tensor_d1,
    uint32_t tile_d0, uint32_t tile_d1, uint64_t stride_d0_elems)
{
  const uint64_t ga = (uint64_t)(uintptr_t)gptr;
  u32x4 g0;
  g0[0] = 1u;                                    // count=1 (valid user descriptor)
  g0[1] = lds_addr;                              // LDS byte address
  g0[2] = (uint32_t)ga;                          // global_addr[31:0]
  g0[3] = (uint32_t)((ga >> 32) & 0x01FFFFFFu)   // global_addr[56:32]
        | (2u << 30);                            // type = 2 ("image")
  i32x8 g1;
  g1[0] = (int)(2u << 16);                       // data_size = 2 -> 4-byte elems
  g1[1] = (int)((tensor_d0 & 0xFFFFu) << 16);    // tensor_dim0[15:0] @ bits 63:48
  g1[2] = (int)(((tensor_d0 >> 16) & 0xFFFFu) |
                ((tensor_d1 & 0xFFFFu) << 16));  // tensor_dim0[31:16] | tensor_dim1[15:0]
  g1[3] = (int)(((tensor_d1 >> 16) & 0xFFFFu) |
                ((tile_d0 & 0xFFFFu) << 16));    // tensor_dim1[31:16] | tile_dim0
  g1[4] = (int)(tile_d1 & 0xFFFFu);              // tile_dim1 (tile_dim2 = 0)
  g1[5] = (int)(uint32_t)(stride_d0_elems & 0xFFFFFFFFull);   // dim0 stride lo
  g1[6] = (int)(uint32_t)((stride_d0_elems >> 32) & 0xFFFFull);
  g1[7] = 0;
  i32x4 z4 = {0, 0, 0, 0};
#if __has_include(<hip/amd_detail/amd_gfx1250_TDM.h>)
  i32x8 z8 = {0, 0, 0, 0, 0, 0, 0, 0};
  __builtin_amdgcn_tensor_load_to_lds(g0, g1, z4, z4, z8, 0);   // clang-23 form
#else
  __builtin_amdgcn_tensor_load_to_lds(g0, g1, z4, z4, 0);       // ROCm 7.2 form
#endif
}

// ---------------------------------------------------------------------------
// Kernel 1: zero numerator + denominator accumulators (d_out is poisoned).
// ---------------------------------------------------------------------------
__global__ __launch_bounds__(256)
void zero_kernel(float4* __restrict__ out, int n4) {
  int i = blockIdx.x * blockDim.x + threadIdx.x;
  const int stride = gridDim.x * blockDim.x;
  const float4 z = make_float4(0.f, 0.f, 0.f, 0.f);
  for (; i < n4; i += stride) out[i] = z;
}

// ---------------------------------------------------------------------------
// Kernel 2: softmax-splat scatter. One 256-thread block per (b, 4-row tile).
// Thread t -> source row y0 + (t>>6), x-quad (t&63)*4: one b128 feat load and
// 16 fp32 global atomics per channel iteration. Flow for the whole tile
// (2 components x 4 rows, planes 65536 elems apart) arrives via a single TDM
// 2D-tile transfer into LDS. Denominator accumulates in the metric region.
// ---------------------------------------------------------------------------
__global__ __launch_bounds__(256)
void splat_kernel(const float* __restrict__ feat,
                  const float* __restrict__ flow,
                  float* __restrict__ num,
                  float* __restrict__ den) {
  __shared__ __align__(16) float flow_lds[2 * ROWS_PER_BLK * NW];   // 8 KB

  const int b  = blockIdx.y;
  const int y0 = blockIdx.x * ROWS_PER_BLK;
  const int tr = threadIdx.x >> 6;           // row within tile: 0..3
  const int tx = (threadIdx.x & 63) * 4;     // x-quad start:    0..252
  const int y  = y0 + tr;

  if (threadIdx.x < 32) {                    // wave 0 issues; TDM ignores EXEC
    const float* gsrc = flow + ((size_t)(b * 2) * NH + y0) * NW;
    tdm_load_rows_to_lds(lds_byte_addr(flow_lds), gsrc,
                         /*tensor_d0=*/ROWS_PER_BLK * NW, /*tensor_d1=*/2,
                         /*tile_d0=*/ROWS_PER_BLK * NW,   /*tile_d1=*/2,
                         /*stride_d0=*/(uint64_t)HWPLANE);
    __builtin_amdgcn_s_wait_tensorcnt(0);
  }
  __syncthreads();

  const float4 fxv = *(const float4*)&flow_lds[tr * NW + tx];
  const float4 fyv = *(const float4*)&flow_lds[ROWS_PER_BLK * NW + tr * NW + tx];
  const float fxa[4] = {fxv.x, fxv.y, fxv.z, fxv.w};
  const float fya[4] = {fyv.x, fyv.y, fyv.z, fyv.w};

  float w[4][4];     // [pixel][corner] bilinear splat weights
  int   o[4][4];     // [pixel][corner] offsets within one (H,W) plane
  bool  v[4][4];     // [pixel][corner] in-bounds flags
#pragma unroll
  for (int j = 0; j < 4; ++j) {
    const float fx = (float)(tx + j) + fxa[j];
    const float fy = (float)y + fya[j];
    const float x0f = floorf(fx);
    const float y0f = floorf(fy);
    const int ix0 = (int)x0f, iy0 = (int)y0f;
    const int ix1 = ix0 + 1,  iy1 = iy0 + 1;
    const float wx1 = fx - x0f, wy1 = fy - y0f;
    const float wx0 = 1.0f - wx1, wy0 = 1.0f - wy1;
    const bool vx0 = (ix0 >= 0) & (ix0 < NW);
    const bool vx1 = (ix1 >= 0) & (ix1 < NW);
    const bool vy0 = (iy0 >= 0) & (iy0 < NH);
    const bool vy1 = (iy1 >= 0) & (iy1 < NH);
    v[j][0] = vx0 & vy0;  o[j][0] = iy0 * NW + ix0;  w[j][0] = wx0 * wy0;
    v[j][1] = vx1 & vy0;  o[j][1] = iy0 * NW + ix1;  w[j][1] = wx1 * wy0;
    v[j][2] = vx0 & vy1;  o[j][2] = iy1 * NW + ix0;  w[j][2] = wx0 * wy1;
    v[j][3] = vx1 & vy1;  o[j][3] = iy1 * NW + ix1;  w[j][3] = wx1 * wy1;
  }

  // Denominator: splat exp(metric)=e through each valid corner.
  float* denp = den + (size_t)b * HWPLANE;
#pragma unroll
  for (int j = 0; j < 4; ++j)
#pragma unroll
    for (int k = 0; k < 4; ++k)
      if (v[j][k]) atomicAdd(denp + o[j][k], kE * w[j][k]);

  // Numerator: one b128 load + up to 16 L2-resident atomics per channel.
  const float* fsrc = feat + ((size_t)(b * NC) * NH + y) * NW + tx;
  float* nbase = num + (size_t)b * NC * HWPLANE;
  for (int c = 0; c < NC; ++c) {
    const int cn = (c + 1 < NC) ? (c + 1) : c;
    __builtin_prefetch(fsrc + (size_t)cn * HWPLANE, 0, 3);   // global_prefetch_b8
    const float4 v4 = *(const float4*)(fsrc + (size_t)c * HWPLANE);
    const float va[4] = {v4.x * kE, v4.y * kE, v4.z * kE, v4.w * kE};
    float* np = nbase + (size_t)c * HWPLANE;
#pragma unroll
    for (int j = 0; j < 4; ++j)
#pragma unroll
      for (int k = 0; k < 4; ++k)
        if (v[j][k]) atomicAdd(np + o[j][k], va[j] * w[j][k]);
  }
}

// ---------------------------------------------------------------------------
// Kernel 3: normalize in place: num[b,c,y,x] /= (den[b,y,x] + EPS).
// b128 loads/stores; four reciprocals live in registers for all 128 channels.
// ---------------------------------------------------------------------------
__global__ __launch_bounds__(256)
void normalize_kernel(float* __restrict__ num, const float* __restrict__ den) {
  const int b  = blockIdx.y;
  const int y  = blockIdx.x * ROWS_PER_BLK + (threadIdx.x >> 6);
  const int tx = (threadIdx.x & 63) * 4;

  const float4 d = *(const float4*)(den + ((size_t)b * NH + y) * NW + tx);
  const float r0 = 1.0f / (d.x + kEPS);
  const float r1 = 1.0f / (d.y + kEPS);
  const float r2 = 1.0f / (d.z + kEPS);
  const float r3 = 1.0f / (d.w + kEPS);

  float* p = num + ((size_t)(b * NC) * NH + y) * NW + tx;
  for (int c = 0; c < NC; ++c) {
    const int cn = (c + 1 < NC) ? (c + 1) : c;
    __builtin_prefetch(p + (size_t)cn * HWPLANE, 1, 3);
    float4* q = (float4*)(p + (size_t)c * HWPLANE);
    float4 t = *q;
    t.x *= r0; t.y *= r1; t.z *= r2; t.w *= r3;
    *q = t;
  }
}

// ---------------------------------------------------------------------------
// Kernel 4: the reference's second output: metric = ones. Must run after
// normalize_kernel (that region doubled as the denominator accumulator).
// ---------------------------------------------------------------------------
__global__ __launch_bounds__(256)
void metric_kernel(float4* __restrict__ m, int n4) {
  const int i = blockIdx.x * blockDim.x + threadIdx.x;
  if (i < n4) m[i] = make_float4(1.f, 1.f, 1.f, 1.f);
}

// ---------------------------------------------------------------------------
extern "C" void kernel_launch(void* const* d_in, const int* in_sizes, int n_in,
                              void* d_out, int out_size, void* d_ws, size_t ws_size,
                              hipStream_t stream) {
  (void)in_sizes; (void)n_in; (void)out_size; (void)d_ws; (void)ws_size;

  const float* feat = (const float*)d_in[0];   // (8,128,256,256) f32
  const float* flow = (const float*)d_in[1];   // (8,2,256,256)   f32
  float* num = (float*)d_out;                  // warped region
  float* den = num + NUM_ELEMS;                // metric region doubles as denominator

  // 1) zero numerator + denominator (67,633,152 floats = 16,908,288 float4)
  const int n4 = (int)((NUM_ELEMS + DEN_ELEMS) / 4);
  zero_kernel<<<4096, 256, 0, stream>>>((float4*)d_out, n4);

  // 2) scatter splat: one block per (b, 4-row tile)
  splat_kernel<<<dim3(NH / ROWS_PER_BLK, NB), 256, 0, stream>>>(feat, flow, num, den);

  // 3) normalize in place (b128 streams)
  normalize_kernel<<<dim3(NH / ROWS_PER_BLK, NB), 256, 0, stream>>>(num, den);

  // 4) metric output = 1.0
  metric_kernel<<<(int)(DEN_ELEMS / 4 / 256), 256, 0, stream>>>((float4*)den,
                                                                (int)(DEN_ELEMS / 4));
}